// HungarianMatcher_64166811402941
// MI455X (gfx1250) — compile-verified
//
#include <hip/hip_runtime.h>
#include <hip/hip_bf16.h>

// Problem constants (from reference)
#define BS_   16
#define Q_    256
#define NC_   2
#define NT_   16
#define P_    72
#define NQ_   (BS_ * Q_)        // 4096 queries
#define T_    (BS_ * NT_)       // 256 targets
#define TROW_ (3 + 2 * P_)      // 147 floats per target row

typedef __attribute__((ext_vector_type(2))) float v2f;
typedef __attribute__((ext_vector_type(8))) float v8f;

// ---------------------------------------------------------------------------
// Kernel 1: per-target preprocessing.
//  - valid counts -> total + per_row -> w[t] = sqrt(total/per_row)/max(...)
//  - extract lower/upper targets, labels
//  - transpose points into interleaved XY[p*T + t] = (x, y) for coalesced
//    broadcast reads in the main kernel
// ---------------------------------------------------------------------------
__global__ __launch_bounds__(T_) void hm_prep_kernel(
    const float* __restrict__ targets,      // (T, 147)
    const int*   __restrict__ tgt_labels,   // (T,)
    float2*      __restrict__ XY,           // (P, T) interleaved x,y
    float*       __restrict__ wout,         // (T,)
    float*       __restrict__ lowers,       // (T,)
    float*       __restrict__ uppers,       // (T,)
    int*         __restrict__ labels)       // (T,)
{
    const int t = threadIdx.x;
    const float* row = targets + t * TROW_;

    int cnt = 0;
    for (int p = 0; p < P_; ++p) {
        float x = row[3 + p];
        float y = row[3 + P_ + p];
        XY[p * T_ + t] = make_float2(x, y);
        cnt += (x >= 0.0f) ? 1 : 0;
    }
    float per_row = (float)cnt;

    __shared__ float red[T_];
    red[t] = per_row;
    __syncthreads();
    for (int s = T_ / 2; s > 0; s >>= 1) {
        if (t < s) red[t] += red[t + s];
        __syncthreads();
    }
    float total = red[0];
    __syncthreads();

    float wraw = sqrtf(total / per_row);
    red[t] = wraw;
    __syncthreads();
    for (int s = T_ / 2; s > 0; s >>= 1) {
        if (t < s) red[t] = fmaxf(red[t], red[t + s]);
        __syncthreads();
    }
    float wmax = red[0];

    wout[t]   = wraw / wmax;
    lowers[t] = row[1];
    uppers[t] = row[2];
    labels[t] = tgt_labels[t];
}

// ---------------------------------------------------------------------------
// Kernel 2: cost matrix. One wave handles one 16(n) x 16(t) output tile.
// Lane L: tl = L&15 selects the target column, half = L>>4 selects which
// 8 query rows this lane accumulates (matching the WMMA f32 C/D layout:
// VGPR j, lanes 0-15 -> M=j, lanes 16-31 -> M=j+8).
// Class cost = -out_prob[n, tid[t]] computed as a WMMA 16x16x4 f32 matmul
// A(16x2 probs, K=2..3 zero) x B(2x16 one-hot labels) -> D tile that drops
// straight into acc[j] with no cross-lane shuffles.
// ---------------------------------------------------------------------------
__global__ __launch_bounds__(256) void hm_cost_kernel(
    const float*  __restrict__ logits,   // (NQ, 2)
    const float*  __restrict__ curves,   // (NQ, 8)
    const float2* __restrict__ XY,       // (P, T)
    const float*  __restrict__ w,        // (T,)
    const float*  __restrict__ lowers,   // (T,)
    const float*  __restrict__ uppers,   // (T,)
    const int*    __restrict__ labels,   // (T,)
    float*        __restrict__ out)      // (NQ, T)
{
    const int lane  = threadIdx.x & 31;
    const int wave  = threadIdx.x >> 5;
    const int tile  = blockIdx.x * 8 + wave;   // 0..4095
    const int ttile = tile & 15;               // 16 target tiles
    const int ntile = tile >> 4;               // 256 query tiles
    const int tl    = lane & 15;
    const int half  = lane >> 4;
    const int tt    = ttile * 16 + tl;         // this lane's target column
    const int nbase = ntile * 16 + half * 8;   // first of this lane's 8 query rows

    // Register-resident coefficients for 8 queries (a,b,c, e, d-f, lower, upper)
    float Ac[8], Bc[8], Cc[8], Ec[8], DFc[8], Low[8], Up[8], acc[8];
#pragma unroll
    for (int j = 0; j < 8; ++j) {
        const float4* cb4 = (const float4*)(curves + (size_t)(nbase + j) * 8);
        float4 lo4 = cb4[0];   // cb0=lower, cb1=upper, cb2=a, cb3=b
        float4 hi4 = cb4[1];   // cb4=c, cb5=d, cb6=e, cb7=f
        Low[j] = lo4.x;  Up[j] = lo4.y;
        Ac[j]  = lo4.z;  Bc[j] = lo4.w;
        Cc[j]  = hi4.x;  Ec[j] = hi4.z;
        DFc[j] = hi4.y - hi4.w;       // d - f
        acc[j] = 0.0f;
    }

    // Main point loop: 1 float2 load feeds 8 query evaluations.
    // out_x = r*(a*r + c) + (e*y + (d-f)),  r = 1/(y-b)
#pragma unroll 2
    for (int p = 0; p < P_; ++p) {
        float2 xy = XY[p * T_ + tt];
        const float x = xy.x, y = xy.y;
        const bool valid = (x >= 0.0f);
#pragma unroll
        for (int j = 0; j < 8; ++j) {
            float dy = y - Bc[j];
            float r  = __builtin_amdgcn_rcpf(dy);
            float ox = fmaf(r, fmaf(Ac[j], r, Cc[j]), fmaf(Ec[j], y, DFc[j]));
            float d  = fabsf(x - ox);
            acc[j] += valid ? d : 0.0f;
        }
    }

    // --- Class cost tile via WMMA (D[m,t] = sum_c prob[m,c]*onehot[c,t]) ---
    // A layout (32-bit A 16x4, ISA-documented): lanes 0-15 hold K=0 (vgpr0),
    // K=1 (vgpr1); lanes 16-31 hold K=2,3 -> zero (K dim only 2 live).
    const int nA = ntile * 16 + tl;
    float l0 = logits[(size_t)nA * 2 + 0];
    float l1 = logits[(size_t)nA * 2 + 1];
    float e10 = __expf(l1 - l0);
    float pr0 = 1.0f / (1.0f + e10);   // softmax class-0 prob
    float pr1 = 1.0f - pr0;            // softmax class-1 prob

    const int lab = labels[tt];
    v2f Aop, Bop;
    Aop.x = (half == 0) ? pr0 : 0.0f;
    Aop.y = (half == 0) ? pr1 : 0.0f;
    Bop.x = (half == 0) ? ((lab == 0) ? 1.0f : 0.0f) : 0.0f;
    Bop.y = (half == 0) ? ((lab == 1) ? 1.0f : 0.0f) : 0.0f;

    v8f Cz = {};
    v8f D = __builtin_amdgcn_wmma_f32_16x16x4_f32(
        /*neg_a=*/false, Aop, /*neg_b=*/false, Bop,
        /*c_mod=*/(short)0, Cz, /*reuse_a=*/false, /*reuse_b=*/false);

    // --- Combine and store (coalesced over t within each lane half) ---
    const float wt = w[tt];
    const float Lt = lowers[tt];
    const float Ut = uppers[tt];
#pragma unroll
    for (int j = 0; j < 8; ++j) {
        const int n = nbase + j;
        float val = fmaf(wt, acc[j], fabsf(Low[j] - Lt) + fabsf(Up[j] - Ut)) - D[j];
        out[(size_t)n * T_ + tt] = val;
    }
}

// ---------------------------------------------------------------------------
// Launcher
// ---------------------------------------------------------------------------
extern "C" void kernel_launch(void* const* d_in, const int* in_sizes, int n_in,
                              void* d_out, int out_size, void* d_ws, size_t ws_size,
                              hipStream_t stream) {
    const float* logits     = (const float*)d_in[0];   // (16,256,2)  f32
    const float* curves     = (const float*)d_in[1];   // (16,256,8)  f32
    const float* targets    = (const float*)d_in[2];   // (16,16,147) f32
    const int*   tgt_labels = (const int*)d_in[3];     // (16,16)     i32
    float* out = (float*)d_out;                        // (16,256,256) f32

    // Workspace layout (~150 KB)
    char* ws = (char*)d_ws;
    float2* XY   = (float2*)ws;                 // P_*T_ float2  = 147456 B
    float* wvec  = (float*)(XY + P_ * T_);      // T_ floats
    float* lows  = wvec + T_;
    float* ups   = lows + T_;
    int*   labs  = (int*)(ups + T_);

    hm_prep_kernel<<<1, T_, 0, stream>>>(targets, tgt_labels, XY, wvec, lows, ups, labs);

    // 4096 tiles (256 n-tiles x 16 t-tiles), 8 waves/block -> 512 blocks
    hm_cost_kernel<<<512, 256, 0, stream>>>(logits, curves, XY, wvec, lows, ups, labs, out);
}